// SpatialEmbLoss_2d_26482768347151
// MI455X (gfx1250) — compile-verified
//
#include <hip/hip_runtime.h>
#include <stdint.h>
#include <math.h>

#define HW   262144   // 512*512
#define NI   16
#define NB   4

typedef __attribute__((ext_vector_type(2))) float v2f;
typedef __attribute__((ext_vector_type(8))) float v8f;

// ---------------- deterministic block helpers ----------------

__device__ __forceinline__ float block_reduce_256(float v, volatile float* buf, int t)
{
    buf[t] = v; __syncthreads();
    #pragma unroll
    for (int off = 128; off > 0; off >>= 1) {
        if (t < off) buf[t] += buf[t + off];
        __syncthreads();
    }
    float r = buf[0];
    __syncthreads();
    return r;
}

__device__ __forceinline__ int block_scan_incl_256(int v, volatile int* buf, int t)
{
    buf[t] = v; __syncthreads();
    for (int off = 1; off < 256; off <<= 1) {
        int x = (t >= off) ? buf[t - off] : 0;
        __syncthreads();
        buf[t] += x;
        __syncthreads();
    }
    return buf[t];   // inclusive; buf[255] = total (stable until next write, fenced)
}

// ---------------- stage 1: per-(b,i) masked stats via WMMA ----------------
// stats[m][n] = sum_p onehot(inst==m+1)[p] * statval(n,p) : a 16xP * Px8 matmul.
// Accumulated with V_WMMA_F32_16X16X4_F32 (K=4 pixels per issue), f32-exact.
// B entries built branchlessly from per-lane 0/1 weights (bit-exact selection).

__global__ __launch_bounds__(256) void k_stats_wmma(
    const float* __restrict__ pred, const int* __restrict__ inst,
    const int* __restrict__ cimg, float* __restrict__ sp)
{
    const int b = blockIdx.y, blk = blockIdx.x, t = threadIdx.x;
    const int wave = t >> 5, lane = t & 31;
    __shared__ int   si[2048];
    __shared__ float sc[2048];
    __shared__ float s0[2048];
    __shared__ float s1[2048];
    const int pix0 = blk * 2048;
    const float* g0 = pred + ((size_t)(b * 5 + 2)) * HW;
    const float* g1 = pred + ((size_t)(b * 5 + 3)) * HW;
    const int* gi = inst + (size_t)b * HW;
    const int* gc = cimg + (size_t)b * HW;
    for (int j = t; j < 2048; j += 256) {
        int p = pix0 + j;
        si[j] = gi[p];
        sc[j] = (gc[p] > 0) ? 1.0f : 0.0f;
        s0[j] = g0[p];
        s1[j] = g1[p];
    }
    __syncthreads();

    const int m  = lane & 15;            // A: instance row;  B: stat column (n==m)
    const int k0 = (lane >> 4) << 1;     // lanes 0-15: K=0,1 ; lanes 16-31: K=2,3
    const float inv = 1.0f / 1023.0f;    // linspace(0,1,1024) step

    // per-lane 0/1 selection weights for the 8 stat columns (m>=8 -> all zero)
    const float f0 = (m == 0) ? 1.0f : 0.0f;   // cnt
    const float f1 = (m == 1) ? 1.0f : 0.0f;   // cm count
    const float f2 = (m == 2) ? 1.0f : 0.0f;   // center_cm x
    const float f3 = (m == 3) ? 1.0f : 0.0f;   // center_cm y
    const float f4 = (m == 4) ? 1.0f : 0.0f;   // mean x num
    const float f5 = (m == 5) ? 1.0f : 0.0f;   // mean y num
    const float f6 = (m == 6) ? 1.0f : 0.0f;   // sigma0 sum
    const float f7 = (m == 7) ? 1.0f : 0.0f;   // sigma1 sum

    v8f c = {};
    const int wb = wave << 8;            // 256 pixels per wave
    #pragma unroll 4
    for (int it = 0; it < 64; ++it) {
        int pa = wb + (it << 2) + k0;
        int pb = pa + 1;
        int ia = si[pa], ib = si[pb];
        v2f a, bm;
        a.x = (ia == m + 1) ? 1.0f : 0.0f;
        a.y = (ib == m + 1) ? 1.0f : 0.0f;
        int gpa = pix0 + pa, gpb = pix0 + pb;
        float xa = (float)(gpa & 511) * inv, ya = (float)(gpa >> 9) * inv;
        float xb = (float)(gpb & 511) * inv, yb = (float)(gpb >> 9) * inv;
        float ca = sc[pa],  cb = sc[pb];
        float za = s0[pa],  zb = s0[pb];
        float wa = s1[pa],  wbv = s1[pb];
        // branchless stat value: exactly one weight is 1 (or none for m>=8)
        float ta = fmaf(f2, xa, fmaf(f3, ya, f1));   // cimg-gated part
        float tb = fmaf(f2, xb, fmaf(f3, yb, f1));
        float va = fmaf(ca, ta, f0);
        float vb = fmaf(cb, tb, f0);
        va = fmaf(f4, xa, va);  vb = fmaf(f4, xb, vb);
        va = fmaf(f5, ya, va);  vb = fmaf(f5, yb, vb);
        va = fmaf(f6, za, va);  vb = fmaf(f6, zb, vb);
        va = fmaf(f7, wa, va);  vb = fmaf(f7, wbv, vb);
        bm.x = va; bm.y = vb;
        c = __builtin_amdgcn_wmma_f32_16x16x4_f32(false, a, false, bm,
                                                  (short)0, c, false, false);
    }
    // C/D layout: lane<16 -> N=lane, M=v ; lane>=16 -> N=lane-16, M=v+8
    int N = lane & 15, hi = lane >> 4;
    if (N < 8) {
        float* out = sp + ((((size_t)b * 128 + blk) * 8 + wave) * 16) * 8;
        #pragma unroll
        for (int v = 0; v < 8; ++v) {
            int M = v + hi * 8;
            out[(size_t)M * 8 + N] = c[v];
        }
    }
}

// stats layout (stride 12): cx, cy, exp(10*s0), exp(10*s1), s0, s1, cntf, cnt, z
__global__ void k_stats_finalize(const float* __restrict__ sp, float* __restrict__ st)
{
    int t = threadIdx.x;
    if (t >= 64) return;
    int b = t >> 4, i = t & 15;
    float q[8] = {0, 0, 0, 0, 0, 0, 0, 0};
    for (int j = 0; j < 1024; ++j) {
        const float* p = sp + ((((size_t)b * 128 + (j >> 3)) * 8 + (j & 7)) * 16 + i) * 8;
        #pragma unroll
        for (int n = 0; n < 8; ++n) q[n] += p[n];
    }
    float cnt  = q[0];
    float cntf = fmaxf(cnt, 1.0f);
    bool  one  = (q[1] == 1.0f);
    float cx   = one ? q[2] : q[4] / cntf;
    float cy   = one ? q[3] : q[5] / cntf;
    float sm0  = q[6] / cntf, sm1 = q[7] / cntf;
    float* o = st + (size_t)(b * 16 + i) * 12;
    o[0] = cx; o[1] = cy;
    o[2] = expf(10.0f * sm0); o[3] = expf(10.0f * sm1);
    o[4] = sm0; o[5] = sm1; o[6] = cntf; o[7] = cnt;
    o[8] = (cnt > 0.0f) ? 1.0f : 0.0f;
}

// ---------------- stage 2: per-pixel pass (one image) ----------------
// Emits sort keys for all 16 instances + deterministic scalar partials.

__global__ __launch_bounds__(256) void k_pixel_pass(
    int b, const float* __restrict__ pred, const int* __restrict__ inst,
    const int* __restrict__ lab, const float* __restrict__ st,
    uint32_t* __restrict__ keys, float* __restrict__ scal)
{
    __shared__ float rbuf[256];
    int t = threadIdx.x, blk = blockIdx.x;
    int pix = blk * 256 + t;
    const float* pb = pred + (size_t)b * 5 * HW;
    float p0  = pb[pix];
    float p1  = pb[(size_t)HW + pix];
    float sg0 = pb[2 * (size_t)HW + pix];
    float sg1 = pb[3 * (size_t)HW + pix];
    float p4  = pb[4 * (size_t)HW + pix];
    int i0 = inst[(size_t)b * HW + pix];
    int lb = lab[(size_t)b * HW + pix];
    const float invc = 1.0f / 1023.0f;
    float x = (float)(pix & 511) * invc, y = (float)(pix >> 9) * invc;
    float sex = tanhf(p0) + x, sey = tanhf(p1) + y;
    float seed = 1.0f / (1.0f + expf(-p4));
    float vvar = 0.0f, sfg = 0.0f, sbg = 0.0f;
    if (lb == 0) sbg = seed * seed;
    #pragma unroll 1
    for (int i = 0; i < NI; ++i) {
        const float* s = st + (size_t)(b * 16 + i) * 12;
        float dx = sex - s[0], dy = sey - s[1];
        float d = expf(-(dx * dx * s[2] + dy * dy * s[3]));
        int g = (i0 == i + 1) ? 1 : 0;
        float e = g ? (2.0f - 2.0f * d) : (2.0f * d);   // = 1 - (2d-1)(2g-1), e in [0,2]
        uint32_t bits = __float_as_uint(e);             // <= 0x40000000
        keys[(size_t)i * HW + pix] = ~((bits << 1) | (uint32_t)g);
        if (g) {
            float t0 = seed - d;
            sfg += t0 * t0;                              // seed_l / FG_WEIGHT
            float d0 = sg0 - s[4], d1 = sg1 - s[5];
            vvar += (d0 * d0 + d1 * d1) * (0.5f / s[6]); // var_l term (N_SIGMA=2)
        }
    }
    float r0 = block_reduce_256(vvar, rbuf, t);
    float r1 = block_reduce_256(sfg,  rbuf, t);
    float r2 = block_reduce_256(sbg,  rbuf, t);
    if (t == 0) {
        float* o = scal + (size_t)(b * 1024 + blk) * 4;
        o[0] = r0; o[1] = r1; o[2] = r2;
    }
}

// ---------------- stage 3: stable segmented LSD radix sort (ascending) ----------------

__global__ __launch_bounds__(256) void k_radix_hist(const uint32_t* __restrict__ src,
                                                    uint32_t* __restrict__ hist, int shift)
{
    __shared__ unsigned h[256];
    int seg = blockIdx.y, blk = blockIdx.x, t = threadIdx.x;
    h[t] = 0; __syncthreads();
    const uint32_t* s = src + (size_t)seg * HW + (size_t)blk * 512;
    unsigned d0 = (s[t]       >> shift) & 255u;
    unsigned d1 = (s[t + 256] >> shift) & 255u;
    atomicAdd(&h[d0], 1u);
    atomicAdd(&h[d1], 1u);
    __syncthreads();
    hist[((size_t)seg * 256 + t) * 512 + blk] = h[t];
}

__global__ __launch_bounds__(256) void k_radix_scan(uint32_t* __restrict__ hist)
{
    __shared__ unsigned tot[256];
    int seg = blockIdx.x, d = threadIdx.x;
    uint32_t* p = hist + ((size_t)seg * 256 + d) * 512;
    unsigned run = 0;
    for (int blk = 0; blk < 512; ++blk) { unsigned v = p[blk]; p[blk] = run; run += v; }
    tot[d] = run; __syncthreads();
    if (d == 0) {
        unsigned acc = 0;
        for (int j = 0; j < 256; ++j) { unsigned v = tot[j]; tot[j] = acc; acc += v; }
    }
    __syncthreads();
    unsigned base = tot[d];
    for (int blk = 0; blk < 512; ++blk) p[blk] += base;
}

__global__ __launch_bounds__(256) void k_radix_scatter(const uint32_t* __restrict__ src,
                                                       uint32_t* __restrict__ dst,
                                                       const uint32_t* __restrict__ hist,
                                                       int shift)
{
    __shared__ uint32_t kb[2][512];
    __shared__ int sbuf[256];
    __shared__ int dstart[256];
    int seg = blockIdx.y, blk = blockIdx.x, t = threadIdx.x;
    const uint32_t* s = src + (size_t)seg * HW + (size_t)blk * 512;
    kb[0][t] = s[t]; kb[0][t + 256] = s[t + 256];
    int cur = 0;
    for (int bit = shift; bit < shift + 8; ++bit) {       // 8 stable 1-bit splits
        __syncthreads();
        uint32_t v0 = kb[cur][2 * t], v1 = kb[cur][2 * t + 1];
        int f0 = (int)((v0 >> bit) & 1u), f1 = (int)((v1 >> bit) & 1u);
        int zc = (1 - f0) + (1 - f1);
        int incl = block_scan_incl_256(zc, sbuf, t);
        int Z = incl - zc;                // zeros before this pair
        int T = sbuf[255];                // total zeros
        int i1 = 2 * t + 1;
        int pos0 = f0 ? (T + 2 * t - Z) : Z;
        int zb1  = Z + (1 - f0);
        int pos1 = f1 ? (T + i1 - zb1) : zb1;
        kb[cur ^ 1][pos0] = v0;
        kb[cur ^ 1][pos1] = v1;
        cur ^= 1;
    }
    __syncthreads();
    for (int e = t; e < 512; e += 256) {
        unsigned d = (kb[cur][e] >> shift) & 255u;
        if (e == 0 || (((kb[cur][e - 1] >> shift) & 255u) != d)) dstart[d] = e;
    }
    __syncthreads();
    for (int e = t; e < 512; e += 256) {
        uint32_t key = kb[cur][e];
        unsigned d = (key >> shift) & 255u;
        unsigned lr = (unsigned)(e - dstart[d]);
        unsigned gp = hist[((size_t)seg * 256 + d) * 512 + blk] + lr;
        dst[(size_t)seg * HW + gp] = key;
    }
}

// ---------------- stage 4: segmented Lovász from sorted keys ----------------

__global__ __launch_bounds__(256) void k_gsum(const uint32_t* __restrict__ keys,
                                              float* __restrict__ gs)
{
    __shared__ float rbuf[256];
    int seg = blockIdx.y, chunk = blockIdx.x, t = threadIdx.x;
    const uint32_t* k = keys + (size_t)seg * HW + (size_t)chunk * 512;
    float v = (float)((1u - (k[t] & 1u)) + (1u - (k[t + 256] & 1u)));  // g = ~key & 1
    float r = block_reduce_256(v, rbuf, t);
    if (t == 0) gs[seg * 512 + chunk] = r;
}

__global__ void k_gscan(float* __restrict__ gs)
{
    if (threadIdx.x) return;
    float* p = gs + (size_t)blockIdx.x * 512;
    float run = 0.0f;                         // exact: counts < 2^24
    for (int j = 0; j < 512; ++j) { float v = p[j]; p[j] = run; run += v; }
}

__device__ __forceinline__ float jacc(float k, float C, float G)
{
    // jaccard at 0-based sorted position k with inclusive positive count C
    return 1.0f - (G - C) / (G + (k + 1.0f) - C);
}

__global__ __launch_bounds__(256) void k_lovasz(
    int b, const uint32_t* __restrict__ keys, const float* __restrict__ gs,
    const float* __restrict__ st, float* __restrict__ ip)
{
    __shared__ int sbuf[256];
    __shared__ float rbuf[256];
    int seg = blockIdx.y, chunk = blockIdx.x, t = threadIdx.x;
    const uint32_t* kk = keys + (size_t)seg * HW + (size_t)chunk * 512;
    uint32_t q0 = ~kk[2 * t], q1 = ~kk[2 * t + 1];
    float e0 = __uint_as_float(q0 >> 1), e1 = __uint_as_float(q1 >> 1);
    int g0 = (int)(q0 & 1u), g1 = (int)(q1 & 1u);
    int incl = block_scan_incl_256(g0 + g1, sbuf, t);
    int excl = incl - (g0 + g1);
    float carry = gs[seg * 512 + chunk];
    float G = st[(size_t)(b * 16 + seg) * 12 + 7];     // gts
    float C0 = carry + (float)(excl + g0);
    float C1 = C0 + (float)g1;
    int kg0 = chunk * 512 + 2 * t;
    float j0  = jacc((float)kg0, C0, G);
    float jm0 = (kg0 > 0) ? jacc((float)(kg0 - 1), C0 - (float)g0, G) : 0.0f;
    float j1  = jacc((float)(kg0 + 1), C1, G);
    float c0 = fmaxf(e0, 0.0f) * (j0 - jm0);
    float c1 = fmaxf(e1, 0.0f) * (j1 - j0);           // C_{k1-1} = C1 - g1 = C0 -> J = j0
    float s = block_reduce_256(c0 + c1, rbuf, t);
    if (t == 0) ip[((size_t)(b * 16 + seg)) * 512 + chunk] = s;
}

// ---------------- stage 5: final combines ----------------

__global__ __launch_bounds__(256) void k_inst_reduce(const float* __restrict__ ip,
                                                     const float* __restrict__ st,
                                                     float* __restrict__ il)
{
    __shared__ float rbuf[256];
    int i = blockIdx.x, b = blockIdx.y, t = threadIdx.x;
    const float* p = ip + ((size_t)(b * 16 + i)) * 512;
    float s = block_reduce_256(p[t] + p[t + 256], rbuf, t);
    if (t == 0) il[b * 16 + i] = s * st[(size_t)(b * 16 + i) * 12 + 8];  // * z
}

__global__ __launch_bounds__(256) void k_scal_reduce(const float* __restrict__ scal,
                                                     float* __restrict__ sr)
{
    __shared__ float rbuf[256];
    int b = blockIdx.x, t = threadIdx.x;
    float a0 = 0, a1 = 0, a2 = 0;
    for (int j = t; j < 1024; j += 256) {
        const float* p = scal + (size_t)(b * 1024 + j) * 4;
        a0 += p[0]; a1 += p[1]; a2 += p[2];
    }
    a0 = block_reduce_256(a0, rbuf, t);
    a1 = block_reduce_256(a1, rbuf, t);
    a2 = block_reduce_256(a2, rbuf, t);
    if (t == 0) { float* o = sr + b * 4; o[0] = a0; o[1] = a1; o[2] = a2; }
}

__global__ void k_final(const float* __restrict__ il, const float* __restrict__ st,
                        const float* __restrict__ sr, float* __restrict__ out)
{
    if (threadIdx.x || blockIdx.x) return;
    float tot = 0.0f;
    for (int b = 0; b < NB; ++b) {
        float obj = 0.0f, instSum = 0.0f;
        for (int i = 0; i < NI; ++i) {
            obj     += st[(size_t)(b * 16 + i) * 12 + 8];
            instSum += il[b * 16 + i];
        }
        float objc = fmaxf(obj, 1.0f);
        float var = sr[b * 4 + 0], sfg = sr[b * 4 + 1], sbg = sr[b * 4 + 2];
        float seed_loss = (sbg + 10.0f * sfg) * (1.0f / (float)HW);  // FG_WEIGHT=10
        tot += instSum / objc + 10.0f * var / objc + seed_loss;      // W_INST=1, W_VAR=10, W_SEED=1
    }
    out[0] = tot * 0.25f;   // mean over B=4
}

// ---------------- host launcher ----------------

extern "C" void kernel_launch(void* const* d_in, const int* in_sizes, int n_in,
                              void* d_out, int out_size, void* d_ws, size_t ws_size,
                              hipStream_t stream)
{
    (void)in_sizes; (void)n_in; (void)out_size; (void)ws_size;
    const float* pred = (const float*)d_in[0];
    const int*   inst = (const int*)d_in[1];
    const int*   lab  = (const int*)d_in[2];
    const int*   cimg = (const int*)d_in[3];
    float* out = (float*)d_out;

    float* w = (float*)d_ws;
    float*    sp    = w;                                   // 4*128*8*16*8 = 524288 f
    float*    st    = sp + 524288;                         // 4*16*12      = 768 f
    float*    scal  = st + 768;                            // 4*1024*4     = 16384 f
    uint32_t* keysA = (uint32_t*)(scal + 16384);           // 16*HW        = 4194304 u32
    uint32_t* keysB = keysA + 4194304;                     // 4194304 u32
    uint32_t* hist  = keysB + 4194304;                     // 16*256*512   = 2097152 u32
    float*    gs    = (float*)(hist + 2097152);            // 16*512       = 8192 f
    float*    ip    = gs + 8192;                           // 4*16*512     = 32768 f
    float*    il    = ip + 32768;                          // 64 f
    float*    sr    = il + 64;                             // 16 f
    // total ~44.3 MB of workspace

    k_stats_wmma<<<dim3(128, 4), 256, 0, stream>>>(pred, inst, cimg, sp);
    k_stats_finalize<<<1, 64, 0, stream>>>(sp, st);

    for (int b = 0; b < NB; ++b) {
        k_pixel_pass<<<1024, 256, 0, stream>>>(b, pred, inst, lab, st, keysA, scal);
        uint32_t* a = keysA; uint32_t* bb = keysB;
        for (int pass = 0; pass < 4; ++pass) {
            int shift = pass * 8;
            k_radix_hist   <<<dim3(512, 16), 256, 0, stream>>>(a, hist, shift);
            k_radix_scan   <<<16,            256, 0, stream>>>(hist);
            k_radix_scatter<<<dim3(512, 16), 256, 0, stream>>>(a, bb, hist, shift);
            uint32_t* tmp = a; a = bb; bb = tmp;
        }
        // after 4 passes the sorted keys are back in keysA (ascending ~packed = descending error)
        k_gsum  <<<dim3(512, 16), 256, 0, stream>>>(keysA, gs);
        k_gscan <<<16, 32, 0, stream>>>(gs);
        k_lovasz<<<dim3(512, 16), 256, 0, stream>>>(b, keysA, gs, st, ip);
    }

    k_inst_reduce<<<dim3(16, 4), 256, 0, stream>>>(ip, st, il);
    k_scal_reduce<<<4, 256, 0, stream>>>(scal, sr);
    k_final<<<1, 1, 0, stream>>>(il, st, sr, out);
}